// GIN_21071109554680
// MI455X (gfx1250) — compile-verified
//
#include <hip/hip_runtime.h>
#include <hip/hip_bf16.h>

typedef __attribute__((ext_vector_type(2))) float v2f;
typedef __attribute__((ext_vector_type(8))) float v8f;

#define D_FEAT 128
#define SCALE 0.1f   // 1/(MAX_NEIGHBORS+1)

// ---------------------------------------------------------------------------
// Kernel 1: agg[i] = (1+eps) * x[i]    (float4 vectorized, memory bound)
// ---------------------------------------------------------------------------
__global__ void gin_init_kernel(const float* __restrict__ x,
                                const float* __restrict__ eps,
                                float* __restrict__ agg, int n4) {
    int i = blockIdx.x * blockDim.x + threadIdx.x;
    if (i >= n4) return;
    float s = 1.0f + eps[0];
    float4 v = ((const float4*)x)[i];
    v.x *= s; v.y *= s; v.z *= s; v.w *= s;
    ((float4*)agg)[i] = v;
}

// ---------------------------------------------------------------------------
// Kernel 2: edge scatter-add. One wave32 per edge: 32 lanes x float4 = 128 f32.
// x (51.2MB) and agg (51.2MB) both fit in the 192MB L2, so gathers and the
// global_atomic_add_f32 stream stay on-chip.
// ---------------------------------------------------------------------------
__global__ void gin_scatter_kernel(const float* __restrict__ x,
                                   const long long* __restrict__ ei,
                                   float* __restrict__ agg, int E) {
    int gtid = blockIdx.x * blockDim.x + threadIdx.x;
    int edge = gtid >> 5;
    int lane = gtid & 31;
    if (edge >= E) return;
    long long src = ei[2LL * edge + 0];
    long long dst = ei[2LL * edge + 1];
    const float4 v = *(const float4*)(x + src * D_FEAT + lane * 4);
    float* p = agg + dst * D_FEAT + lane * 4;
    atomicAdd(p + 0, v.x);
    atomicAdd(p + 1, v.y);
    atomicAdd(p + 2, v.z);
    atomicAdd(p + 3, v.w);
}

// ---------------------------------------------------------------------------
// Kernel 3: out = relu( (agg*SCALE) @ W + b )  via V_WMMA_F32_16X16X4_F32.
// Block = 256 threads = 8 waves; wave w owns output columns [16w,16w+16).
// B fragments (W columns, SCALE pre-folded) are preloaded into 64 VGPRs once
// per block and reused across M-tiles (grid-stride loop over 6250 tiles).
//
// f32 WMMA layouts (ISA 7.12.2):
//   A 16x4 : lane<16 -> A[m][k0..k0+1], lane>=16 -> A[m][k0+2..k0+3]  (v2f)
//   B 4x16 : lane<16 -> B[k0..k0+1][n], lane>=16 -> B[k0+2..k0+3][n]  (v2f)
//   C 16x16: vgpr r  -> row r (lanes 0-15) / row r+8 (lanes 16-31)    (v8f)
// ---------------------------------------------------------------------------
__global__ void gin_gemm_kernel(const float* __restrict__ h,
                                const float* __restrict__ W,
                                const float* __restrict__ bias,
                                float* __restrict__ out, int mtiles) {
    const int wv   = threadIdx.x >> 5;      // wave in block: 0..7
    const int lane = threadIdx.x & 31;
    const int half = lane >> 4;             // 0: K pair (0,1), 1: K pair (2,3)
    const int lm   = lane & 15;
    const int n0   = wv * 16;               // column tile
    const int col  = n0 + lm;

    // Preload B fragments for full K=128 (32 steps of K=4), SCALE folded in.
    v2f bw[32];
    const float* wp = W + (2 * half) * D_FEAT + col;
#pragma unroll
    for (int kk = 0; kk < 32; ++kk) {
        bw[kk].x = wp[(4 * kk + 0) * D_FEAT] * SCALE;
        bw[kk].y = wp[(4 * kk + 1) * D_FEAT] * SCALE;
    }
    const float bias_v = bias[col];

    for (int mt = blockIdx.x; mt < mtiles; mt += gridDim.x) {
        const int m0 = mt * 16;
        const float* ha = h + (long long)(m0 + lm) * D_FEAT + 2 * half;
        v8f acc = {};
#pragma unroll
        for (int kk = 0; kk < 32; ++kk) {
            v2f a = *(const v2f*)(ha + 4 * kk);
            // 8 args: (neg_a, A, neg_b, B, c_mod, C, reuse_a, reuse_b)
            acc = __builtin_amdgcn_wmma_f32_16x16x4_f32(
                false, a, false, bw[kk], (short)0, acc, false, false);
        }
        float* orow = out + (long long)(m0 + 8 * half) * D_FEAT + col;
#pragma unroll
        for (int r = 0; r < 8; ++r) {
            float v = acc[r] + bias_v;
            v = v > 0.0f ? v : 0.0f;
            __builtin_nontemporal_store(v, orow + r * D_FEAT); // keep L2 for x/agg
        }
    }
}

// ---------------------------------------------------------------------------
extern "C" void kernel_launch(void* const* d_in, const int* in_sizes, int n_in,
                              void* d_out, int out_size, void* d_ws, size_t ws_size,
                              hipStream_t stream) {
    const float*     x   = (const float*)d_in[0];
    const long long* ei  = (const long long*)d_in[1];  // int64 (E,2)
    const float*     Wm  = (const float*)d_in[2];      // (128,128)
    const float*     b   = (const float*)d_in[3];      // (128,)
    const float*     eps = (const float*)d_in[4];      // (1,)
    float* out = (float*)d_out;
    float* agg = (float*)d_ws;                          // N*128 f32 = 51.2 MB

    const int Nn = in_sizes[0] / D_FEAT;               // 100000
    const int E  = in_sizes[1] / 2;                    // 3200000

    // 1) agg = (1+eps)*x
    {
        int n4 = (Nn * D_FEAT) / 4;
        gin_init_kernel<<<(n4 + 255) / 256, 256, 0, stream>>>(x, eps, agg, n4);
    }
    // 2) atomic scatter-add of sender features (one wave per edge)
    {
        long long threads = (long long)E * 32;
        int blocks = (int)((threads + 255) / 256);
        gin_scatter_kernel<<<blocks, 256, 0, stream>>>(x, ei, agg, E);
    }
    // 3) WMMA GEMM + bias + relu
    {
        int mtiles = Nn / 16;                          // 6250
        gin_gemm_kernel<<<1024, 256, 0, stream>>>(agg, Wm, b, out, mtiles);
    }
}